// ComposedFeatureTransformer_45861660787506
// MI455X (gfx1250) — compile-verified
//
#include <hip/hip_runtime.h>

// ---------------------------------------------------------------------------
// NNUE feature transformer for MI455X (gfx1250).
//
// Roofline: ~3.2 GB of gathered weight-row reads @ 23.3 TB/s => ~140 us floor;
// compute is ~1.6 GFLOP (irrelevant), so this is a pure gather-bandwidth
// problem. WMMA does not apply (block-diagonal einsum: every batch row
// contracts a different gathered 32x3080 matrix). The CDNA5 path that matters
// is GLOBAL_LOAD_ASYNC_TO_LDS_B128 + s_wait_asynccnt: a triple-buffered
// gather pipeline in LDS that keeps two 24KB row-batches in flight per
// workgroup while the FMA accumulation drains the third.
// ---------------------------------------------------------------------------

#define L1      3072
#define NPSQT   8
#define DDIM    3080          // L1 + NPSQT
#define KACT    32            // MAX_ACTIVE
#define OUTD    6152          // 2*L1 + NPSQT
#define NTH     256           // 8 waves (wave32)
#define F4_L1   768           // L1 / 4 float4 chunks per row
#define NBUF    3             // triple buffering

#if __has_builtin(__builtin_amdgcn_global_load_async_to_lds_b128) && \
    __has_builtin(__builtin_amdgcn_s_wait_asynccnt)
#define FT_ASYNC 1
#pragma message("gfx1250 async-to-LDS builtins AVAILABLE: using async triple-buffered gather")
#else
#define FT_ASYNC 0
#pragma message("gfx1250 async-to-LDS builtins NOT available in this pass: direct-load path")
#endif

typedef int v4i __attribute__((vector_size(4 * sizeof(int))));

__device__ __forceinline__ float4 ld4(const float* p) {
    return *reinterpret_cast<const float4*>(p);
}
__device__ __forceinline__ void st4(float* p, float4 v) {
    *reinterpret_cast<float4*>(p) = v;
}
__device__ __forceinline__ void fma4(float4& acc, float s, float4 v) {
    acc.x = fmaf(s, v.x, acc.x);
    acc.y = fmaf(s, v.y, acc.y);
    acc.z = fmaf(s, v.z, acc.z);
    acc.w = fmaf(s, v.w, acc.w);
}
__device__ __forceinline__ float4 add4(float4 a, float4 b) {
    float4 r; r.x = a.x + b.x; r.y = a.y + b.y; r.z = a.z + b.z; r.w = a.w + b.w;
    return r;
}
__device__ __forceinline__ float clamp01m(float x, float m) {
    return fminf(fmaxf(x, 0.0f), m);
}

__global__ __launch_bounds__(NTH)
void ft_kernel(const int* __restrict__ w_idx, const float* __restrict__ w_val,
               const int* __restrict__ b_idx, const float* __restrict__ b_val,
               const float* __restrict__ us,  const float* __restrict__ them,
               const int* __restrict__ ftmax,
               const float* __restrict__ W,   const float* __restrict__ bias,
               float* __restrict__ out)
{
    // Triple-buffered staging for the L1 part of the two gathered rows.
    __shared__ float4 sW[NBUF][F4_L1];
    __shared__ float4 sB[NBUF][F4_L1];
    __shared__ int    s_wi[KACT];
    __shared__ float  s_wv[KACT];
    __shared__ int    s_bi[KACT];
    __shared__ float  s_bv[KACT];
    __shared__ float  sredW[NTH];
    __shared__ float  sredB[NTH];

    const int row = blockIdx.x;
    const int t   = threadIdx.x;

    if (t < KACT) {
        s_wi[t] = w_idx[row * KACT + t];
        s_wv[t] = w_val[row * KACT + t];
    } else if (t < 2 * KACT) {
        const int u = t - KACT;
        s_bi[u] = b_idx[row * KACT + u];
        s_bv[u] = b_val[row * KACT + u];
    }

    // Hoisted uniform/epilogue data: these overlap the gather pipeline.
    const float usv = us[row];
    const float thv = them[row];
    const float fm  = (float)ftmax[0];
    float4 bi4[3];
#pragma unroll
    for (int j = 0; j < 3; ++j) bi4[j] = ld4(bias + (j * NTH + t) * 4);

    __syncthreads();

    float4 wacc[3], bacc[3];
#pragma unroll
    for (int j = 0; j < 3; ++j) {
        wacc[j] = make_float4(0.f, 0.f, 0.f, 0.f);
        bacc[j] = make_float4(0.f, 0.f, 0.f, 0.f);
    }

#if FT_ASYNC
    // --- async triple-buffered gather pipeline --------------------------
    // Each wave issues exactly 6 full-EXEC async B128 copies per feature k
    // (3 chunks x {w-row, b-row}); async loads retire in order, so
    // s_wait_asynccnt(6) at the top of iteration k means batch k has landed
    // in LDS while batch k+1 stays in flight. Batch k+2 (issued below) goes
    // into buffer (k-1)%3, whose readers all passed this iteration's
    // barrier already -> a single barrier per iteration is sufficient.
    auto issue = [&](int k) {
        const int    buf = k % NBUF;
        const float* wr  = W + (size_t)s_wi[k] * DDIM;
        const float* br  = W + (size_t)s_bi[k] * DDIM;
#pragma unroll
        for (int j = 0; j < 3; ++j) {
            const int c4 = j * NTH + t;
            __builtin_amdgcn_global_load_async_to_lds_b128(
                (__attribute__((address_space(1))) v4i*)(wr + c4 * 4),
                (__attribute__((address_space(3))) v4i*)&sW[buf][c4], 0, 0);
            __builtin_amdgcn_global_load_async_to_lds_b128(
                (__attribute__((address_space(1))) v4i*)(br + c4 * 4),
                (__attribute__((address_space(3))) v4i*)&sB[buf][c4], 0, 0);
        }
    };

    issue(0);
    issue(1);

    for (int k = 0; k < KACT; ++k) {
        if (k + 1 < KACT) __builtin_amdgcn_s_wait_asynccnt(6);
        else              __builtin_amdgcn_s_wait_asynccnt(0);
        __syncthreads();                 // batch k visible; iter k-1 reads done

        if (k + 2 < KACT) issue(k + 2);  // refill (k-1)%3 before computing

        const float wv  = s_wv[k];
        const float bv  = s_bv[k];
        const int   buf = k % NBUF;
#pragma unroll
        for (int j = 0; j < 3; ++j) {
            const int c4 = j * NTH + t;
            fma4(wacc[j], wv, sW[buf][c4]);
            fma4(bacc[j], bv, sB[buf][c4]);
        }
    }
#else
    // --- fallback: direct vector-load gather ----------------------------
#pragma unroll 4
    for (int k = 0; k < KACT; ++k) {
        const float  wv = s_wv[k];
        const float  bv = s_bv[k];
        const float* wr = W + (size_t)s_wi[k] * DDIM;
        const float* br = W + (size_t)s_bi[k] * DDIM;
#pragma unroll
        for (int j = 0; j < 3; ++j) {
            const int c4 = j * NTH + t;
            fma4(wacc[j], wv, ld4(wr + c4 * 4));
            fma4(bacc[j], bv, ld4(br + c4 * 4));
        }
    }
#endif

    // --- epilogue: bias + us/them mix + clamp ---------------------------
    float* orow = out + (size_t)row * OUTD;

#pragma unroll
    for (int j = 0; j < 3; ++j) {
        const int c4 = j * NTH + t;
        const float4 w4 = add4(wacc[j], bi4[j]);
        const float4 b4 = add4(bacc[j], bi4[j]);
        float4 o1, o2;
        o1.x = clamp01m(fmaf(usv, w4.x, thv * b4.x), fm);
        o1.y = clamp01m(fmaf(usv, w4.y, thv * b4.y), fm);
        o1.z = clamp01m(fmaf(usv, w4.z, thv * b4.z), fm);
        o1.w = clamp01m(fmaf(usv, w4.w, thv * b4.w), fm);
        o2.x = clamp01m(fmaf(usv, b4.x, thv * w4.x), fm);
        o2.y = clamp01m(fmaf(usv, b4.y, thv * w4.y), fm);
        o2.z = clamp01m(fmaf(usv, b4.z, thv * w4.z), fm);
        o2.w = clamp01m(fmaf(usv, b4.w, thv * w4.w), fm);
        st4(orow + c4 * 4, o1);
        st4(orow + L1 + c4 * 4, o2);
    }

    // --- PSQT: fully parallel gather (k = t/8, p = t%8) + LDS reduction --
    {
        const int kk = t >> 3;
        const int p  = t & 7;
        sredW[t] = s_wv[kk] * W[(size_t)s_wi[kk] * DDIM + L1 + p];
        sredB[t] = s_bv[kk] * W[(size_t)s_bi[kk] * DDIM + L1 + p];
        __syncthreads();
#pragma unroll
        for (int str = NTH / 2; str >= NPSQT; str >>= 1) {
            if (t < str) {
                sredW[t] += sredW[t + str];
                sredB[t] += sredB[t + str];
            }
            __syncthreads();
        }
        if (t < NPSQT) {
            const float bb = bias[L1 + t];
            orow[2 * L1 + t] = ((sredW[t] + bb) - (sredB[t] + bb)) * (usv - 0.5f);
        }
    }
}

extern "C" void kernel_launch(void* const* d_in, const int* in_sizes, int n_in,
                              void* d_out, int out_size, void* d_ws, size_t ws_size,
                              hipStream_t stream) {
    const int*   w_idx = (const int*)  d_in[0];
    const float* w_val = (const float*)d_in[1];
    const int*   b_idx = (const int*)  d_in[2];
    const float* b_val = (const float*)d_in[3];
    const float* us    = (const float*)d_in[4];
    const float* them  = (const float*)d_in[5];
    const int*   ftmax = (const int*)  d_in[6];
    const float* W     = (const float*)d_in[7];
    const float* bias  = (const float*)d_in[8];
    float* out = (float*)d_out;

    const int batch = in_sizes[0] / KACT;   // 4096
    dim3 grid(batch), block(NTH);
    hipLaunchKernelGGL(ft_kernel, grid, block, 0, stream,
                       w_idx, w_val, b_idx, b_val, us, them, ftmax, W, bias, out);
    (void)d_ws; (void)ws_size; (void)out_size; (void)n_in;
}